// equi_conv4_21603685499533
// MI455X (gfx1250) — compile-verified
//
#include <hip/hip_runtime.h>
#include <math.h>

// ---------------------------------------------------------------------------
// Problem dims (fixed by the reference)
// ---------------------------------------------------------------------------
#define PNT   2048     // P
#define CIN   32
#define DOUT  32
#define NNB   10       // N neighbors
#define AA    12       // A directions
#define K13   13       // kernel slots (top, bot, 10 sym, center)
#define MROWS (DOUT * K13 * 12)   // 4992  (d,k,r) flattened
#define KDIM  (CIN * 3)           // 96    (c,i)   flattened

// ---------------------------------------------------------------------------
// Workspace layout (float offsets)
// ---------------------------------------------------------------------------
#define OFF_VS     0                         // 12*3   = 36
#define OFF_SYMR   36                        // 5*9    = 45
#define OFF_VERT   81                        // 12*9   = 108  -> 189, pad 192
#define OFF_BIASF  192                       // 32*13  = 416  -> 608, pad 640
#define OFF_KDNRO  640                       // 32*12*12*3 = 13824 -> 14464
#define OFF_A      14464                     // 4992*96 = 479232 -> 493696
#define OFF_FMB    493696                    // 96*2048 = 196608 -> 690304
#define OFF_NDN    690304                    // 2048*10*3 = 61440 -> 751744
#define OFF_FO     751744                    // 4992*2048 = 10223616
// total = 10,975,360 floats  (~43.9 MB)

typedef __attribute__((ext_vector_type(2))) float v2f;
typedef __attribute__((ext_vector_type(8))) float v8f;

// ---------------------------------------------------------------------------
// K0: icosahedron constants (double precision like the reference, then f32)
// ---------------------------------------------------------------------------
__global__ void consts_kernel(float* __restrict__ ws) {
  if (threadIdx.x != 0 || blockIdx.x != 0) return;
  const double phi = (1.0 + sqrt(5.0)) * 0.5;
  double v[12][3];
  int idx = 0;
  for (int ai = 0; ai < 2; ++ai) {
    double a = (ai == 0) ? 1.0 : -1.0;
    for (int bi = 0; bi < 2; ++bi) {
      double b = (bi == 0) ? phi : -phi;
      v[idx][0] = 0.0; v[idx][1] = a;   v[idx][2] = b;   ++idx;
      v[idx][0] = a;   v[idx][1] = b;   v[idx][2] = 0.0; ++idx;
      v[idx][0] = b;   v[idx][1] = 0.0; v[idx][2] = a;   ++idx;
    }
  }
  for (int r = 0; r < 12; ++r) {
    double n = sqrt(v[r][0]*v[r][0] + v[r][1]*v[r][1] + v[r][2]*v[r][2]);
    for (int i = 0; i < 3; ++i) {
      v[r][i] /= n;
      ws[OFF_VS + r*3 + i] = (float)v[r][i];
    }
  }
  // symR: 5 rotations about z
  for (int m = 0; m < 5; ++m) {
    double t = 2.0 * 3.14159265358979323846 * (double)m / 5.0;
    double ct = cos(t), st = sin(t);
    double R[3][3] = {{ct, -st, 0.0}, {st, ct, 0.0}, {0.0, 0.0, 1.0}};
    for (int i = 0; i < 3; ++i)
      for (int j = 0; j < 3; ++j)
        ws[OFF_SYMR + m*9 + i*3 + j] = (float)R[i][j];
  }
  // verticeR: Rodrigues rotation z -> vertex r
  for (int r = 0; r < 12; ++r) {
    double ux = v[r][0], uy = v[r][1], uz = v[r][2];
    double c = uz;
    double ax = -uy, ay = ux;                 // cross(z, u)
    double s = sqrt(ax*ax + ay*ay);
    double R[3][3];
    if (s < 1e-9) {
      for (int i = 0; i < 3; ++i) for (int j = 0; j < 3; ++j) R[i][j] = 0.0;
      if (c > 0.0) { R[0][0] = 1.0; R[1][1] = 1.0; R[2][2] = 1.0; }
      else         { R[0][0] = 1.0; R[1][1] = -1.0; R[2][2] = -1.0; }
    } else {
      double kx = ax / s, ky = ay / s;        // kz = 0
      double K[3][3] = {{0.0, 0.0, ky}, {0.0, 0.0, -kx}, {-ky, kx, 0.0}};
      double K2[3][3];
      for (int i = 0; i < 3; ++i)
        for (int j = 0; j < 3; ++j) {
          double acc = 0.0;
          for (int q = 0; q < 3; ++q) acc += K[i][q] * K[q][j];
          K2[i][j] = acc;
        }
      for (int i = 0; i < 3; ++i)
        for (int j = 0; j < 3; ++j)
          R[i][j] = (i == j ? 1.0 : 0.0) + s * K[i][j] + (1.0 - c) * K2[i][j];
    }
    for (int i = 0; i < 3; ++i)
      for (int j = 0; j < 3; ++j)
        ws[OFF_VERT + r*9 + i*3 + j] = (float)R[i][j];
  }
}

// ---------------------------------------------------------------------------
// K1a: build A matrix [4992 x 96]:  A[(d*13+k)*12+r][c*3+i] = W_ro[d,c,k,r,i]
// one thread per (d, c, k13)
// ---------------------------------------------------------------------------
__global__ void prep_wro_kernel(const float* __restrict__ W,
                                const float* __restrict__ ws,
                                float* __restrict__ Amat) {
  int idx = blockIdx.x * blockDim.x + threadIdx.x;
  if (idx >= DOUT * CIN * K13) return;
  int k = idx % K13;
  int c = (idx / K13) % CIN;
  int d = idx / (K13 * CIN);
  const float* wrow = W + (d * CIN + c) * 9;
  float wf[3];
  if (k == 0)       { wf[0] = 0.f; wf[1] = 0.f; wf[2] =  wrow[0]; }
  else if (k == 1)  { wf[0] = 0.f; wf[1] = 0.f; wf[2] = -wrow[1]; }
  else if (k == 12) { wf[0] = 0.f; wf[1] = 0.f; wf[2] =  wrow[8]; }
  else {
    int kk = (k - 2) / 5;
    int m  = (k - 2) % 5;
    float wv0 = wrow[2 + kk*3 + 0];
    float wv1 = wrow[2 + kk*3 + 1];
    float wv2 = wrow[2 + kk*3 + 2];
    const float* S = ws + OFF_SYMR + m * 9;
    for (int i = 0; i < 3; ++i)
      wf[i] = S[i*3+0]*wv0 + S[i*3+1]*wv1 + S[i*3+2]*wv2;
  }
  for (int r = 0; r < 12; ++r) {
    const float* Rr = ws + OFF_VERT + r * 9;
    int mrow = (d * K13 + k) * 12 + r;
    for (int i = 0; i < 3; ++i) {
      float val = Rr[i*3+0]*wf[0] + Rr[i*3+1]*wf[1] + Rr[i*3+2]*wf[2];
      Amat[mrow * KDIM + c*3 + i] = val;
    }
  }
}

// ---------------------------------------------------------------------------
// K1b: kdn_ro[d,k12,r,i]   one thread per (d, k12)
// ---------------------------------------------------------------------------
__global__ void prep_kdnro_kernel(const float* __restrict__ directions,
                                  const float* __restrict__ ws,
                                  float* __restrict__ kdnro) {
  int idx = blockIdx.x * blockDim.x + threadIdx.x;
  if (idx >= DOUT * 12) return;
  int k = idx % 12;
  int d = idx / 12;
  float v[3];
  if (k == 0)      { v[0] = 0.f; v[1] = 0.f; v[2] =  1.f; }
  else if (k == 1) { v[0] = 0.f; v[1] = 0.f; v[2] = -1.f; }
  else {
    int kk = (k - 2) / 5;
    int m  = (k - 2) % 5;
    const float* dv = directions + (d * 2 + kk) * 3;
    float n = sqrtf(dv[0]*dv[0] + dv[1]*dv[1] + dv[2]*dv[2]);
    n = fmaxf(n, 1e-12f);
    float u0 = dv[0]/n, u1 = dv[1]/n, u2 = dv[2]/n;
    const float* S = ws + OFF_SYMR + m * 9;
    for (int i = 0; i < 3; ++i)
      v[i] = S[i*3+0]*u0 + S[i*3+1]*u1 + S[i*3+2]*u2;
  }
  for (int r = 0; r < 12; ++r) {
    const float* Rr = ws + OFF_VERT + r * 9;
    for (int i = 0; i < 3; ++i)
      kdnro[((d*12 + k)*12 + r)*3 + i] =
          Rr[i*3+0]*v[0] + Rr[i*3+1]*v[1] + Rr[i*3+2]*v[2];
  }
}

// ---------------------------------------------------------------------------
// K1c: bias_full[d][k13]
// ---------------------------------------------------------------------------
__global__ void prep_bias_kernel(const float* __restrict__ bias,
                                 float* __restrict__ biasf) {
  int idx = blockIdx.x * blockDim.x + threadIdx.x;
  if (idx >= DOUT * K13) return;
  int k = idx % K13;
  int d = idx / K13;
  int sel;
  if (k == 0) sel = 0;
  else if (k == 1) sel = 1;
  else if (k <= 6) sel = 2;
  else if (k <= 11) sel = 3;
  else sel = 4;
  biasf[idx] = bias[d * 5 + sel];
}

// ---------------------------------------------------------------------------
// K2: attention softmax + fm.   one wave per point p; lane = channel c.
// writes fm (part of d_out) and the GEMM B matrix [96 x 2048] in scratch.
// ---------------------------------------------------------------------------
__global__ void __launch_bounds__(256)
atten_fm_kernel(const float* __restrict__ fmap, const float* __restrict__ fcw,
                const float* __restrict__ ws, float* __restrict__ out_fm,
                float* __restrict__ fmB) {
  __shared__ float xs[8][AA][33];
  const int lane = threadIdx.x & 31;
  const int w    = threadIdx.x >> 5;
  const int p    = blockIdx.x * 8 + w;
  const int c    = lane;

  float x[AA];
  const float* src = fmap + (c * PNT + p) * AA;
  #pragma unroll
  for (int a = 0; a < AA; ++a) { x[a] = src[a]; xs[w][a][c] = x[a]; }
  __syncthreads();

  float l[AA];
  const float* fr = fcw + c * CIN;   // row c of fc_w  (logit channel c)
  #pragma unroll
  for (int a = 0; a < AA; ++a) {
    float s = 0.f;
    #pragma unroll
    for (int cc = 0; cc < CIN; ++cc) s += xs[w][a][cc] * fr[cc];
    l[a] = s;
  }
  float mx = l[0];
  #pragma unroll
  for (int a = 1; a < AA; ++a) mx = fmaxf(mx, l[a]);
  float se = 0.f;
  #pragma unroll
  for (int a = 0; a < AA; ++a) { l[a] = expf(l[a] - mx); se += l[a]; }
  float inv = 1.f / se;
  float y[AA];
  #pragma unroll
  for (int a = 0; a < AA; ++a) y[a] = l[a] * inv * x[a];

  const float* vs = ws + OFF_VS;
  #pragma unroll
  for (int r = 0; r < 3; ++r) {
    float f = 0.f;
    #pragma unroll
    for (int a = 0; a < AA; ++a) f += y[a] * vs[a*3 + r];
    out_fm[(c * PNT + p) * 3 + r] = f;          // fm output (b,c,p,r)
    fmB[(c*3 + r) * PNT + p] = f;               // GEMM B [96 x 2048]
  }
}

// ---------------------------------------------------------------------------
// K3: normalized neighbor directions ndn[p][n][3]
// ---------------------------------------------------------------------------
__global__ void ndn_kernel(const int* __restrict__ nbidx,
                           const float* __restrict__ vert,
                           float* __restrict__ ndn) {
  int t = blockIdx.x * blockDim.x + threadIdx.x;
  if (t >= PNT * NNB) return;
  int p = t / NNB;
  int q = nbidx[t];
  float dx = vert[q*3+0] - vert[p*3+0];
  float dy = vert[q*3+1] - vert[p*3+1];
  float dz = vert[q*3+2] - vert[p*3+2];
  float n = sqrtf(dx*dx + dy*dy + dz*dz);
  n = fmaxf(n, 1e-12f);
  float inv = 1.f / n;
  ndn[t*3+0] = dx * inv;
  ndn[t*3+1] = dy * inv;
  ndn[t*3+2] = dz * inv;
}

// ---------------------------------------------------------------------------
// K4: WMMA f32 GEMM:  fo[4992 x 2048] = A[4992 x 96] * B[96 x 2048] + bias
// grid (8, 312), 128 threads (4 waves). Each wave: one 16(M) x 64(N) strip,
// 24 K-steps of V_WMMA_F32_16X16X4_F32 per accumulator (96 WMMAs / wave).
// f32 A/B fragment layout: 2 VGPRs/lane; lanes 0-15 K={0,1}, lanes 16-31 K={2,3}.
// ---------------------------------------------------------------------------
__global__ void __launch_bounds__(128)
gemm_fo_kernel(const float* __restrict__ Amat, const float* __restrict__ Bm,
               const float* __restrict__ biasf, float* __restrict__ fo) {
  const int lane  = threadIdx.x & 31;
  const int wave  = threadIdx.x >> 5;
  const int mbase = blockIdx.y * 16;
  const int nbase = blockIdx.x * 256 + wave * 64;
  const int mr    = lane & 15;            // A row / B,C column within tile
  const int kh    = (lane >> 4) * 2;      // K sub-offset per half-wave

  v8f acc[4];
  #pragma unroll
  for (int t = 0; t < 4; ++t)
    #pragma unroll
    for (int v = 0; v < 8; ++v) acc[t][v] = 0.f;

  const float* arow = Amat + (mbase + mr) * KDIM + kh;
  for (int k0 = 0; k0 < KDIM; k0 += 4) {
    v2f af;
    af.x = arow[k0];
    af.y = arow[k0 + 1];
    #pragma unroll
    for (int t = 0; t < 4; ++t) {
      const float* bp = Bm + (k0 + kh) * PNT + nbase + t * 16 + mr;
      v2f bf;
      bf.x = bp[0];
      bf.y = bp[PNT];
      acc[t] = __builtin_amdgcn_wmma_f32_16x16x4_f32(
          /*neg_a=*/false, af, /*neg_b=*/false, bf,
          /*c_mod=*/(short)0, acc[t], /*reuse_a=*/false, /*reuse_b=*/false);
    }
  }

  const int rbase = (lane >> 4) * 8;      // C layout: VGPR v -> M = v (+8 hi half)
  #pragma unroll
  for (int t = 0; t < 4; ++t) {
    int col = nbase + t * 16 + mr;
    #pragma unroll
    for (int v = 0; v < 8; ++v) {
      int m = mbase + rbase + v;
      fo[m * PNT + col] = acc[t][v] + biasf[m / 12];
    }
  }
}

// ---------------------------------------------------------------------------
// K5: out[d,p,r] = sum_k max_n( fo[(d,k,r), nb(p,n)] * relu(<kdn_ro, ndn>) )
//                  + fo[(d,12,r), p]
// one thread per output element (d * 2048 * 12)
// ---------------------------------------------------------------------------
__global__ void __launch_bounds__(256)
final_kernel(const float* __restrict__ fo, const float* __restrict__ kdnro,
             const float* __restrict__ ndn, const int* __restrict__ nbidx,
             float* __restrict__ out) {
  int t = blockIdx.x * blockDim.x + threadIdx.x;
  if (t >= DOUT * PNT * 12) return;
  int r = t % 12;
  int p = (t / 12) % PNT;
  int d = t / (12 * PNT);

  int   nb[NNB];
  float nv[NNB][3];
  #pragma unroll
  for (int n = 0; n < NNB; ++n) {
    nb[n] = nbidx[p * NNB + n];
    nv[n][0] = ndn[(p * NNB + n) * 3 + 0];
    nv[n][1] = ndn[(p * NNB + n) * 3 + 1];
    nv[n][2] = ndn[(p * NNB + n) * 3 + 2];
  }

  float sum = 0.f;
  for (int k = 0; k < 12; ++k) {
    const float* kd = kdnro + ((d * 12 + k) * 12 + r) * 3;
    float k0 = kd[0], k1 = kd[1], k2 = kd[2];
    const float* forow = fo + ((d * K13 + k) * 12 + r) * PNT;
    float mx = -3.402823466e+38f;
    #pragma unroll
    for (int n = 0; n < NNB; ++n) {
      float theta = k0 * nv[n][0] + k1 * nv[n][1] + k2 * nv[n][2];
      theta = fmaxf(theta, 0.f);
      float g = forow[nb[n]];
      mx = fmaxf(mx, g * theta);
    }
    sum += mx;
  }
  float center = fo[((d * K13 + 12) * 12 + r) * PNT + p];
  out[t] = sum + center;
}

// ---------------------------------------------------------------------------
// Launch
// ---------------------------------------------------------------------------
extern "C" void kernel_launch(void* const* d_in, const int* in_sizes, int n_in,
                              void* d_out, int out_size, void* d_ws, size_t ws_size,
                              hipStream_t stream) {
  const int*   nbidx      = (const int*)  d_in[0];   // (1, 2048, 10)
  const float* vertices   = (const float*)d_in[1];   // (1, 2048, 3)
  const float* fmap       = (const float*)d_in[2];   // (1, 32, 2048, 12)
  const float* W          = (const float*)d_in[3];   // (32, 32, 9)
  const float* bias       = (const float*)d_in[4];   // (32, 5)
  const float* directions = (const float*)d_in[5];   // (32, 2, 3)
  const float* fcw        = (const float*)d_in[6];   // (32, 32)

  float* out   = (float*)d_out;                      // out (786432) ++ fm (196608)
  float* wsf   = (float*)d_ws;

  float* ws_consts = wsf;                 // VS / SYMR / VERTICER
  float* biasf     = wsf + OFF_BIASF;
  float* kdnro     = wsf + OFF_KDNRO;
  float* Amat      = wsf + OFF_A;
  float* fmB       = wsf + OFF_FMB;
  float* ndn       = wsf + OFF_NDN;
  float* fo        = wsf + OFF_FO;
  float* out_fm    = out + DOUT * PNT * 12;

  consts_kernel<<<1, 1, 0, stream>>>(ws_consts);

  {
    int n = DOUT * CIN * K13;                       // 13312
    prep_wro_kernel<<<(n + 255) / 256, 256, 0, stream>>>(W, ws_consts, Amat);
  }
  {
    int n = DOUT * 12;                              // 384
    prep_kdnro_kernel<<<(n + 255) / 256, 256, 0, stream>>>(directions, ws_consts, kdnro);
  }
  {
    int n = DOUT * K13;                             // 416
    prep_bias_kernel<<<(n + 255) / 256, 256, 0, stream>>>(bias, biasf);
  }

  atten_fm_kernel<<<PNT / 8, 256, 0, stream>>>(fmap, fcw, ws_consts, out_fm, fmB);

  {
    int n = PNT * NNB;                              // 20480
    ndn_kernel<<<(n + 255) / 256, 256, 0, stream>>>(nbidx, vertices, ndn);
  }

  {
    dim3 grid(PNT / 256, MROWS / 16);               // (8, 312)
    gemm_fo_kernel<<<grid, 128, 0, stream>>>(Amat, fmB, biasf, fo);
  }

  {
    int n = DOUT * PNT * 12;                        // 786432
    final_kernel<<<(n + 255) / 256, 256, 0, stream>>>(fo, kdnro, ndn, nbidx, out);
  }
}